// ValenceConstraintLayer_33887291965648
// MI455X (gfx1250) — compile-verified
//
#include <hip/hip_runtime.h>
#include <hip/hip_bf16.h>
#include <stdint.h>

// MAX_VALENCES table from the reference; default 4.0 outside [0,11)
__constant__ float c_maxval[11] = {4.f, 3.f, 2.f, 6.f, 1.f, 1.f, 1.f, 1.f, 4.f, 4.f, 4.f};

// clang-native vector type (what the async-LDS builtins expect), not HIP_vector_type
typedef int v4i __attribute__((vector_size(16)));
typedef __attribute__((address_space(1))) v4i global_v4i;
typedef __attribute__((address_space(3))) v4i lds_v4i;

// ---------------- K0: zero the bond counters ----------------
__global__ void zero_counts_kernel(unsigned* __restrict__ counts, int n) {
    int i = blockIdx.x * blockDim.x + threadIdx.x;
    if (i < n) counts[i] = 0u;
}

// ---------------- K1: scatter-add of ones over edge sources ----------------
// counts[] is 2MB for N=500k -> lives in L2 (192MB); atomics are L2-side RMW.
__global__ void __launch_bounds__(256)
scatter_bonds_kernel(const int* __restrict__ row, long long E,
                     unsigned* __restrict__ counts) {
    const long long stride = (long long)gridDim.x * blockDim.x;
    const long long i = (long long)blockIdx.x * blockDim.x + threadIdx.x;
    const long long Ev = E >> 2;              // int4-vectorized count
    const int4* __restrict__ row4 = (const int4*)row;
    for (long long v = i; v < Ev; v += stride) {
        // speculative prefetch of the next index chunk (global_prefetch_b8)
        __builtin_prefetch(row4 + v + stride, 0, 3);
        int4 r = row4[v];
        atomicAdd(&counts[r.x], 1u);
        atomicAdd(&counts[r.y], 1u);
        atomicAdd(&counts[r.z], 1u);
        atomicAdd(&counts[r.w], 1u);
    }
    if (i == 0) {                              // tail (E % 4)
        for (long long e = (Ev << 2); e < E; ++e) atomicAdd(&counts[row[e]], 1u);
    }
}

// ---------------- K2 (fast path, D==128): scale rows + per-block v^2 partials ----
// One wave32 per node row: 32 lanes x float4 = 128 floats. Each wave async-stages
// its own 512B row into LDS (GLOBAL_LOAD_ASYNC_TO_LDS_B128 / ASYNCcnt), then waits
// on its own asynccnt -> no cross-wave dependency on the staged data.
__global__ void __launch_bounds__(256)
scale_h128_kernel(const float* __restrict__ h,
                  const unsigned* __restrict__ counts,
                  const int* __restrict__ atom_types,
                  float* __restrict__ h_out,
                  float* __restrict__ partials, int N) {
    __shared__ float tile[8 * 128];   // 8 rows x 128 f32 = 4KB
    __shared__ float psum[8];

    const int t    = threadIdx.x;
    const int warp = t >> 5;
    const int lane = t & 31;
    const int node0 = blockIdx.x * 8;
    const long long gbase = (long long)blockIdx.x * (8 * 128);

    if (node0 + 7 < N) {
        // all 8 rows valid: full-block async stage, 16B per lane
#if __has_builtin(__builtin_amdgcn_global_load_async_to_lds_b128)
        __builtin_amdgcn_global_load_async_to_lds_b128(
            (global_v4i*)(h + gbase + t * 4), (lds_v4i*)&tile[t * 4], 0, 0);
#if __has_builtin(__builtin_amdgcn_s_wait_asynccnt)
        __builtin_amdgcn_s_wait_asynccnt(0);
#else
        asm volatile("s_wait_asynccnt 0" ::: "memory");
#endif
#else
        float4 x = *(const float4*)(h + gbase + t * 4);
        *(float4*)&tile[t * 4] = x;
#endif
    } else {
        int node = node0 + warp;
        if (node < N) {
            float4 x = *(const float4*)&h[(long long)node * 128 + lane * 4];
            *(float4*)&tile[warp * 128 + lane * 4] = x;
        }
    }
    __syncthreads();

    const int node = node0 + warp;
    if (node < N) {
        float cnt = (float)counts[node];
        int   ty  = atom_types[node];
        float ma  = (ty >= 0 && ty < 11) ? c_maxval[ty] : 4.0f;
        float v   = fmaxf(cnt - ma, 0.0f);
        float s   = 1.0f - 0.1f * v;
        float4 x  = *(const float4*)&tile[warp * 128 + lane * 4];
        x.x *= s; x.y *= s; x.z *= s; x.w *= s;
        *(float4*)&h_out[(long long)node * 128 + lane * 4] = x;
        if (lane == 0) psum[warp] = v * v;
    } else if (lane == 0) {
        psum[warp] = 0.0f;
    }
    __syncthreads();
    if (t == 0) {   // fixed-order partial -> deterministic
        float ssum = 0.0f;
        #pragma unroll
        for (int i = 0; i < 8; ++i) ssum += psum[i];
        partials[blockIdx.x] = ssum;
    }
}

// ---------------- K2 fallback (generic D) ----------------
__global__ void __launch_bounds__(256)
violation_kernel(const unsigned* __restrict__ counts,
                 const int* __restrict__ atom_types,
                 float* __restrict__ vio, float* __restrict__ partials, int N) {
    __shared__ float sdata[256];
    int i = blockIdx.x * 256 + threadIdx.x;
    float v = 0.0f;
    if (i < N) {
        float cnt = (float)counts[i];
        int   ty  = atom_types[i];
        float ma  = (ty >= 0 && ty < 11) ? c_maxval[ty] : 4.0f;
        v = fmaxf(cnt - ma, 0.0f);
        vio[i] = v;
    }
    sdata[threadIdx.x] = v * v;
    __syncthreads();
    for (int s = 128; s > 0; s >>= 1) {
        if (threadIdx.x < s) sdata[threadIdx.x] += sdata[threadIdx.x + s];
        __syncthreads();
    }
    if (threadIdx.x == 0) partials[blockIdx.x] = sdata[0];
}

__global__ void scale_generic_kernel(const float* __restrict__ h,
                                     const float* __restrict__ vio,
                                     float* __restrict__ out,
                                     long long total, int D) {
    long long i = (long long)blockIdx.x * blockDim.x + threadIdx.x;
    long long stride = (long long)gridDim.x * blockDim.x;
    for (; i < total; i += stride) {
        long long n = i / D;
        out[i] = h[i] * (1.0f - 0.1f * vio[n]);
    }
}

// ---------------- K3: deterministic final reduction ----------------
__global__ void __launch_bounds__(256)
reduce_partials_kernel(const float* __restrict__ partials, int n,
                       float* __restrict__ out, float invN) {
    __shared__ float sdata[256];
    float local = 0.0f;
    for (int i = threadIdx.x; i < n; i += 256) local += partials[i];
    sdata[threadIdx.x] = local;
    __syncthreads();
    for (int s = 128; s > 0; s >>= 1) {
        if (threadIdx.x < s) sdata[threadIdx.x] += sdata[threadIdx.x + s];
        __syncthreads();
    }
    if (threadIdx.x == 0) *out = sdata[0] * invN;
}

extern "C" void kernel_launch(void* const* d_in, const int* in_sizes, int n_in,
                              void* d_out, int out_size, void* d_ws, size_t ws_size,
                              hipStream_t stream) {
    const float* h          = (const float*)d_in[0];
    const int*   edge_index = (const int*)d_in[1];   // row = first E entries
    const int*   atom_types = (const int*)d_in[3];

    const int N = in_sizes[3];
    if (N <= 0) return;
    const int D = in_sizes[0] / N;                    // 128 in reference
    const long long E = (long long)in_sizes[1] / 2;

    float* h_out          = (float*)d_out;
    float* violations_out = (float*)d_out + (size_t)N * (size_t)D;

    // workspace layout: counts u32[N] | partials f32[maxBlocks] | vio f32[N]
    unsigned* counts = (unsigned*)d_ws;
    size_t counts_bytes = (((size_t)N * 4) + 255) & ~(size_t)255;
    int nblocks_fast = (N + 7) / 8;
    int nblocks_gen  = (N + 255) / 256;
    int maxblocks    = (nblocks_fast > nblocks_gen) ? nblocks_fast : nblocks_gen;
    size_t part_bytes = (((size_t)maxblocks * 4) + 255) & ~(size_t)255;
    float* partials = (float*)((char*)d_ws + counts_bytes);
    float* vio      = (float*)((char*)d_ws + counts_bytes + part_bytes);

    zero_counts_kernel<<<(N + 255) / 256, 256, 0, stream>>>(counts, N);

    {
        long long Ev = E >> 2;
        long long b  = (Ev + 255) / 256;
        int blocks = (b < 1) ? 1 : ((b > 65535) ? 65535 : (int)b);
        scatter_bonds_kernel<<<blocks, 256, 0, stream>>>(edge_index, E, counts);
    }

    int npart;
    if (D == 128) {
        npart = nblocks_fast;
        scale_h128_kernel<<<npart, 256, 0, stream>>>(h, counts, atom_types,
                                                     h_out, partials, N);
    } else {
        npart = nblocks_gen;
        violation_kernel<<<npart, 256, 0, stream>>>(counts, atom_types, vio,
                                                    partials, N);
        long long total = (long long)N * D;
        long long b = (total + 255) / 256;
        int blocks = (b > 65535) ? 65535 : (int)b;
        scale_generic_kernel<<<blocks, 256, 0, stream>>>(h, vio, h_out, total, D);
    }

    reduce_partials_kernel<<<1, 256, 0, stream>>>(partials, npart, violations_out,
                                                  1.0f / (float)N);
}